// MyNetRGCN_83958020702634
// MI455X (gfx1250) — compile-verified
//
#include <hip/hip_runtime.h>
#include <hip/hip_bf16.h>

typedef __attribute__((ext_vector_type(2))) float v2f;
typedef __attribute__((ext_vector_type(8))) float v8f;

#define BB 32
#define NN 512
#define DD 512
#define RR 8
#define EE 4096
#define KTOT (9 * DD) /* 4608 */
#define NGLOB 15000
#define NSENSE 25000

// ---------------------------------------------------------------------------
// Kernel 1: per-sample edge scan + neighbor-sum folding.
// Builds Z[b, 0..8, 0..511] (self row + per-relation normalized neighbor sums)
// and biasacc[b, d] = bias0[0,d] + sum_r bias_{r+1}[0,d]/c_r.
// Deterministic: mask built with idempotent atomicOr, then iterated in fixed
// word/bit order by every thread (uniform control flow, coalesced x loads).
// ---------------------------------------------------------------------------
__global__ __launch_bounds__(512) void rgcn_prep_kernel(
    const float* __restrict__ x, const int* __restrict__ edge_index,
    const int* __restrict__ edge_type, const float* __restrict__ biasr,
    float* __restrict__ Z, float* __restrict__ biasacc) {
  __shared__ unsigned mask[RR * 16];  // 512 bits per relation
  const int b = blockIdx.x;
  const int tid = threadIdx.x;
  if (tid < RR * 16) mask[tid] = 0u;
  __syncthreads();

  const int* ei = edge_index + (size_t)b * 2 * EE;
  const int* et = edge_type + (size_t)b * EE;
  for (int e = tid; e < EE; e += 512) {
    if (ei[e] == 0) {  // source node 0
      int dst = ei[EE + e];
      int r = et[e];
      atomicOr(&mask[r * 16 + (dst >> 5)], 1u << (dst & 31));
    }
  }
  __syncthreads();

  const int d = tid;  // feature dim, 512 threads
  float selfv = x[((size_t)b * NN + 0) * DD + d];
  Z[(size_t)b * KTOT + d] = selfv;
  float bacc = biasr[d];  // biasr_all[0][0][d]
  for (int r = 0; r < RR; ++r) {
    float acc = 0.0f;
    int cnt = 0;
    for (int wi = 0; wi < 16; ++wi) {
      unsigned w = mask[r * 16 + wi];
      cnt += __popc(w);
      while (w) {
        int bit = __ffs(w) - 1;
        w &= w - 1;
        int j = (wi << 5) + bit;
        acc += x[((size_t)b * NN + j) * DD + d];
      }
    }
    float inv = 1.0f / (float)(cnt > 0 ? cnt : 1);
    Z[(size_t)b * KTOT + (size_t)(r + 1) * DD + d] = acc * inv;
    bacc += biasr[(size_t)(r + 1) * NN * DD + d] * inv;
  }
  biasacc[(size_t)b * DD + d] = bacc;
}

// ---------------------------------------------------------------------------
// Kernel 2: h0 = relu(Z[32,4608] @ Wr_flat[4608,512] + biasacc), f32 WMMA.
// 64 tiles of 16x16 -> 8 blocks x 8 waves, one tile per wave, K chained
// through the WMMA f32 accumulator (16x16x4 per instruction).
// ---------------------------------------------------------------------------
__global__ __launch_bounds__(256) void rgcn_gemm_kernel(
    const float* __restrict__ Z, const float* __restrict__ Wr,
    const float* __restrict__ biasacc, float* __restrict__ h0) {
  const int wave = threadIdx.x >> 5;
  const int lane = threadIdx.x & 31;
  const int tile = blockIdx.x * 8 + wave;  // 0..63
  const int m0 = (tile >> 5) * 16;         // 2 M-tiles
  const int n0 = (tile & 31) * 16;         // 32 N-tiles
  const int half = lane >> 4;              // K-half within WMMA
  const int l = lane & 15;

  const float* arow = Z + (size_t)(m0 + l) * KTOT + half * 2;
  const float* bcol = Wr + (size_t)(half * 2) * DD + (n0 + l);

  v8f acc = {};
  for (int k0 = 0; k0 < KTOT; k0 += 4) {
    v2f a = *(const v2f*)(arow + k0);  // A[m, k0+2*half .. +1], 8B aligned
    v2f bv;
    bv.x = bcol[(size_t)k0 * DD];        // B[k0+2*half,   n]
    bv.y = bcol[(size_t)(k0 + 1) * DD];  // B[k0+2*half+1, n]
    acc = __builtin_amdgcn_wmma_f32_16x16x4_f32(
        false, a, false, bv, (short)0, acc, false, false);
  }
  const int col = n0 + l;
  for (int v = 0; v < 8; ++v) {
    int row = m0 + v + half * 8;  // f32 C/D layout: M = v + 8*(lane>=16)
    float val = acc[v] + biasacc[(size_t)row * DD + col];
    h0[(size_t)row * DD + col] = val > 0.0f ? val : 0.0f;
  }
}

// ---------------------------------------------------------------------------
// Kernel 3: logits = h0[32,512] @ W[N,512]^T + b, written straight to d_out.
// Bandwidth-bound on W (82 MB total across both heads): each W row is read
// exactly once as contiguous v2f (b64) per lane; adjacent waves share the
// same B tile across the two M tiles for L2 reuse.
// ---------------------------------------------------------------------------
__global__ __launch_bounds__(256) void head_gemm_kernel(
    const float* __restrict__ h0, const float* __restrict__ W,
    const float* __restrict__ bvec, int Nout, float* __restrict__ out) {
  const int wave = threadIdx.x >> 5;
  const int lane = threadIdx.x & 31;
  const int ntiles = (Nout + 15) >> 4;
  const int t = blockIdx.x * 8 + wave;
  if (t >= 2 * ntiles) return;  // wave-uniform
  const int m0 = (t & 1) * 16;
  const int n0 = (t >> 1) * 16;
  const int half = lane >> 4;
  const int l = lane & 15;

  int nrow = n0 + l;
  if (nrow > Nout - 1) nrow = Nout - 1;  // clamp loads; store is guarded
  const float* arow = h0 + (size_t)(m0 + l) * DD + half * 2;
  const float* brow = W + (size_t)nrow * DD + half * 2;

  v8f acc = {};
#pragma unroll 4
  for (int k0 = 0; k0 < DD; k0 += 4) {
    v2f a = *(const v2f*)(arow + k0);
    v2f b = *(const v2f*)(brow + k0);
    acc = __builtin_amdgcn_wmma_f32_16x16x4_f32(
        false, a, false, b, (short)0, acc, false, false);
  }
  const int col = n0 + l;
  if (col < Nout) {
    float bb = bvec[col];
    for (int v = 0; v < 8; ++v) {
      int row = m0 + v + half * 8;
      out[(size_t)row * Nout + col] = acc[v] + bb;
    }
  }
}

// ---------------------------------------------------------------------------
// Kernel 4: per-row max + log(sum(exp)) over the logits in d_out.
// Rows 0..31 = global head (len 15000), rows 32..63 = sense head (len 25000).
// ---------------------------------------------------------------------------
__global__ __launch_bounds__(256) void lsm_stats_kernel(
    const float* __restrict__ out, float* __restrict__ stats) {
  __shared__ float red[256];
  const int row = blockIdx.x;
  const float* p;
  int len;
  if (row < BB) {
    p = out + (size_t)row * NGLOB;
    len = NGLOB;
  } else {
    p = out + (size_t)BB * NGLOB + (size_t)(row - BB) * NSENSE;
    len = NSENSE;
  }
  const int tid = threadIdx.x;
  float m = -3.4e38f;
  for (int i = tid; i < len; i += 256) m = fmaxf(m, p[i]);
  red[tid] = m;
  __syncthreads();
  for (int s = 128; s > 0; s >>= 1) {
    if (tid < s) red[tid] = fmaxf(red[tid], red[tid + s]);
    __syncthreads();
  }
  m = red[0];
  __syncthreads();
  float sum = 0.0f;
  for (int i = tid; i < len; i += 256) sum += expf(p[i] - m);
  red[tid] = sum;
  __syncthreads();
  for (int s = 128; s > 0; s >>= 1) {
    if (tid < s) red[tid] += red[tid + s];
    __syncthreads();
  }
  if (tid == 0) {
    stats[2 * row] = m;
    stats[2 * row + 1] = logf(red[0]);
  }
}

// Kernel 5: in-place log-softmax apply: out[i] -= (max + log(sumexp)) of row.
__global__ __launch_bounds__(256) void lsm_apply_kernel(
    float* __restrict__ out, const float* __restrict__ stats) {
  int i = blockIdx.x * 256 + threadIdx.x;
  if (i >= BB * NGLOB + BB * NSENSE) return;
  int row;
  if (i < BB * NGLOB)
    row = i / NGLOB;
  else
    row = BB + (i - BB * NGLOB) / NSENSE;
  out[i] -= stats[2 * row] + stats[2 * row + 1];
}

extern "C" void kernel_launch(void* const* d_in, const int* in_sizes, int n_in,
                              void* d_out, int out_size, void* d_ws,
                              size_t ws_size, hipStream_t stream) {
  const float* x = (const float*)d_in[0];
  const int* edge_index = (const int*)d_in[1];
  const int* edge_type = (const int*)d_in[2];
  const float* Wr_all = (const float*)d_in[3];
  const float* biasr_all = (const float*)d_in[4];
  const float* W_global = (const float*)d_in[5];
  const float* b_global = (const float*)d_in[6];
  const float* W_sense = (const float*)d_in[7];
  const float* b_sense = (const float*)d_in[8];
  float* out = (float*)d_out;

  float* ws = (float*)d_ws;
  float* Z = ws;                       // 32*4608 floats
  float* biasacc = Z + BB * KTOT;      // 32*512
  float* h0 = biasacc + BB * DD;       // 32*512
  float* stats = h0 + BB * DD;         // 128

  rgcn_prep_kernel<<<BB, 512, 0, stream>>>(x, edge_index, edge_type, biasr_all,
                                           Z, biasacc);
  rgcn_gemm_kernel<<<8, 256, 0, stream>>>(Z, Wr_all, biasacc, h0);

  {
    int tiles = 2 * ((NGLOB + 15) / 16);
    head_gemm_kernel<<<(tiles + 7) / 8, 256, 0, stream>>>(h0, W_global,
                                                          b_global, NGLOB, out);
  }
  {
    int tiles = 2 * ((NSENSE + 15) / 16);
    head_gemm_kernel<<<(tiles + 7) / 8, 256, 0, stream>>>(
        h0, W_sense, b_sense, NSENSE, out + (size_t)BB * NGLOB);
  }
  lsm_stats_kernel<<<2 * BB, 256, 0, stream>>>(out, stats);
  {
    int total = BB * NGLOB + BB * NSENSE;
    lsm_apply_kernel<<<(total + 255) / 256, 256, 0, stream>>>(out, stats);
  }
}